// DMTetGeometry_61692910240025
// MI455X (gfx1250) — compile-verified
//
#include <hip/hip_runtime.h>
#include <stdint.h>
#include <math.h>

#define TPB 256      // tet/scan/uv kernels
#define ZB  128      // edge kernels: threads along z (requires n <= ZB; n=97)

// ---- marching-tets tables (match reference) ----
__constant__ int c_TRI[16][6] = {
  {-1,-1,-1,-1,-1,-1},{1,0,2,-1,-1,-1},{4,0,3,-1,-1,-1},{1,4,2,1,3,4},
  {3,1,5,-1,-1,-1},{2,3,0,2,5,3},{1,4,0,1,5,4},{4,2,5,-1,-1,-1},
  {4,5,2,-1,-1,-1},{4,1,0,4,5,1},{3,2,0,3,5,2},{1,3,5,-1,-1,-1},
  {4,1,2,4,3,1},{3,0,4,-1,-1,-1},{2,0,1,-1,-1,-1},{-1,-1,-1,-1,-1,-1}};
__constant__ int c_NT[16] = {0,1,1,2,1,2,2,1,1,2,2,1,2,1,1,0};
__constant__ int c_EC[6][2] = {{0,1},{0,2},{0,3},{1,2},{1,3},{2,3}};

// Edge slot e = v*7 + k; direction bits d = k+1: dx=d>>2, dy=(d>>1)&1, dz=d&1.
// Offsets ascend with k (1 < n < n+1 < n^2 < ...) => slot order == jnp.unique sort.

// wave32 inclusive scan + 4-wave LDS combine; returns exclusive prefix of cnt,
// leaves per-wave totals in ws[0..3].
__device__ __forceinline__ unsigned block_excl_scan_zb(unsigned cnt, unsigned* ws) {
  unsigned lane = threadIdx.x & 31u, wid = threadIdx.x >> 5;
  unsigned incl = cnt;
  #pragma unroll
  for (int d = 1; d < 32; d <<= 1) {
    unsigned t = __shfl_up(incl, d, 32);
    if (lane >= (unsigned)d) incl += t;
  }
  if (lane == 31) ws[wid] = incl;
  __syncthreads();
  unsigned woff = 0;
  for (unsigned i = 0; i < wid; ++i) woff += ws[i];
  return woff + incl - cnt;
}

// Async-stage the 4 sdf z-columns this (x,y) block needs into LDS.
// col c: (x + (c>>1), y + (c&1)), clamped to lattice; coalesced in z.
__device__ __forceinline__ void stage_columns(const float* __restrict__ sdf,
                                              int x, int y, int n,
                                              float (*col)[ZB]) {
  int z = threadIdx.x;
  int zc = z < n ? z : n - 1;
  #pragma unroll
  for (int c = 0; c < 4; ++c) {
    int xc = x + (c >> 1); if (xc >= n) xc = n - 1;
    int yc = y + (c & 1);  if (yc >= n) yc = n - 1;
    const float* g = sdf + ((size_t)((size_t)xc * n + yc) * n + zc);
    unsigned lds = (unsigned)(uintptr_t)(void*)&col[c][z];  // low 32b = LDS offset
    asm volatile("global_load_async_to_lds_b32 %0, %1, off" :: "v"(lds), "v"(g) : "memory");
  }
  asm volatile("s_wait_asynccnt 0" ::: "memory");
  __syncthreads();
}

// Cube-corner sdf from LDS columns; returns 7-bit crossing mask (bit k = d-1).
__device__ __forceinline__ unsigned edge_mask_lds(const float (*col)[ZB], int z,
                                                  int bx, int by, int bz,
                                                  float* s /*[8]*/) {
  int z1 = (z + 1 < ZB) ? z + 1 : z;
  #pragma unroll
  for (int d = 0; d < 8; ++d)
    s[d] = col[d >> 1][(d & 1) ? z1 : z];
  int vld[8] = {1, bz, by, by & bz, bx, bx & bz, bx & by, bx & by & bz};
  int o0 = s[0] >= 0.f;
  unsigned m = 0;
  #pragma unroll
  for (int d = 1; d < 8; ++d) {
    int o = s[d] >= 0.f;
    m |= (unsigned)(vld[d] & (o ^ o0)) << (d - 1);
  }
  return m;
}

// ---------------- Pass 1a: count crossing edges per (x,y)-block -------------
__global__ void edge_classify(const float* __restrict__ sdf, int n,
                              unsigned* __restrict__ blockSums) {
  __shared__ float col[4][ZB];
  int y = blockIdx.x, x = blockIdx.y, z = threadIdx.x;
  stage_columns(sdf, x, y, n, col);
  unsigned cnt = 0;
  if (z < n) {
    float s[8];
    unsigned m = edge_mask_lds(col, z, x + 1 < n, y + 1 < n, z + 1 < n, s);
    cnt = (unsigned)__popc(m);
  }
  __shared__ unsigned ws[ZB / 32];
  (void)block_excl_scan_zb(cnt, ws);
  __syncthreads();
  if (threadIdx.x == 0) {
    unsigned s2 = 0;
    for (int i = 0; i < ZB / 32; ++i) s2 += ws[i];
    blockSums[blockIdx.y * gridDim.x + blockIdx.x] = s2;  // x*n + y
  }
}

// ---------------- generic single-block tiled exclusive scan -----------------
__global__ void scan_exclusive(const unsigned* __restrict__ in,
                               unsigned* __restrict__ out,
                               unsigned* __restrict__ total, int B) {
  __shared__ unsigned tile[TPB];
  __shared__ unsigned carry;
  if (threadIdx.x == 0) carry = 0;
  __syncthreads();
  for (int base = 0; base < B; base += TPB) {
    int i = base + threadIdx.x;
    unsigned v = (i < B) ? in[i] : 0u;
    tile[threadIdx.x] = v;
    __syncthreads();
    for (int off = 1; off < TPB; off <<= 1) {
      unsigned add = (threadIdx.x >= (unsigned)off) ? tile[threadIdx.x - off] : 0u;
      __syncthreads();
      tile[threadIdx.x] += add;
      __syncthreads();
    }
    unsigned excl = tile[threadIdx.x] - v;
    if (i < B) out[i] = carry + excl;
    __syncthreads();
    if (threadIdx.x == TPB - 1) carry += tile[TPB - 1];
    __syncthreads();
  }
  if (threadIdx.x == 0) *total = carry;
}

// ---------------- Pass 2a: classify tets (async global->LDS staging) --------
__global__ void tet_classify(const int* __restrict__ tet, const float* __restrict__ sdf,
                             int F, unsigned char* __restrict__ code,
                             unsigned* __restrict__ sums1, unsigned* __restrict__ sums2) {
  __shared__ int4 st[TPB];
  int t  = blockIdx.x * TPB + threadIdx.x;
  int tc = (t < F) ? t : (F - 1);                 // clamp: keep EXEC uniform
  const int4* gp = (const int4*)tet + tc;
  unsigned lds = (unsigned)(uintptr_t)(void*)&st[threadIdx.x];
  asm volatile("global_load_async_to_lds_b128 %0, %1, off" :: "v"(lds), "v"(gp) : "memory");
  {
    int pf = tc + TPB; if (pf >= F) pf = F - 1;
    __builtin_prefetch((const int4*)tet + pf, 0, 3);   // global_prefetch_b8
  }
  asm volatile("s_wait_asynccnt 0" ::: "memory");
  __syncthreads();
  int4 vt = st[threadIdx.x];
  int c = (sdf[vt.x] >= 0.f ? 1 : 0) | (sdf[vt.y] >= 0.f ? 2 : 0) |
          (sdf[vt.z] >= 0.f ? 4 : 0) | (sdf[vt.w] >= 0.f ? 8 : 0);
  int nt = c_NT[c];
  int m1 = (t < F) && (nt == 1);
  int m2 = (t < F) && (nt == 2);
  if (t < F) code[t] = (unsigned char)c;
  unsigned long long b1 = __ballot(m1);
  unsigned long long b2 = __ballot(m2);
  __shared__ unsigned w1[TPB / 32], w2[TPB / 32];
  if ((threadIdx.x & 31) == 0) {
    w1[threadIdx.x >> 5] = (unsigned)__popcll(b1);
    w2[threadIdx.x >> 5] = (unsigned)__popcll(b2);
  }
  __syncthreads();
  if (threadIdx.x == 0) {
    unsigned a = 0, b = 0;
    for (int i = 0; i < TPB / 32; ++i) { a += w1[i]; b += w2[i]; }
    sums1[blockIdx.x] = a;
    sums2[blockIdx.x] = b;
  }
}

// ---------------- Pass 1b: emit interpolated vertices + edge map ------------
__global__ void edge_emit(const float* __restrict__ sdf, const float* __restrict__ pos,
                          int n, const unsigned* __restrict__ blockOff,
                          int* __restrict__ edgeMap, float* __restrict__ outVerts) {
  __shared__ float col[4][ZB];
  int y = blockIdx.x, x = blockIdx.y, z = threadIdx.x;
  int n2 = n * n;
  stage_columns(sdf, x, y, n, col);
  int active = z < n;
  int v = active ? (x * n + y) * n + z : 0;
  float s[8];
  unsigned m = 0;
  if (active)
    m = edge_mask_lds(col, z, x + 1 < n, y + 1 < n, z + 1 < n, s);
  unsigned cnt = (unsigned)__popc(m);
  __shared__ unsigned ws[ZB / 32];
  unsigned excl = block_excl_scan_zb(cnt, ws);
  unsigned idx = blockOff[blockIdx.y * gridDim.x + blockIdx.x] + excl;
  if (active) {
    const int offs[7] = {1, n, n + 1, n2, n2 + 1, n2 + n, n2 + n + 1};
    #pragma unroll
    for (int k = 0; k < 7; ++k) {
      if (m & (1u << k)) {
        int vb = v + offs[k];
        float sa = s[0], sb = s[k + 1];
        float denom = sa - sb;                   // e_sdf=[sa,-sb]; sum
        float w0 = -sb / denom, w1 = sa / denom; // flipped / denom
        #pragma unroll
        for (int j = 0; j < 6; ++j)
          outVerts[(size_t)idx * 6 + j] =
              pos[(size_t)v * 6 + j] * w0 + pos[(size_t)vb * 6 + j] * w1;
        edgeMap[v * 7 + k] = (int)idx;
        ++idx;
      } else {
        edgeMap[v * 7 + k] = -1;
      }
    }
  }
}

// ---------------- Pass 2b: emit faces + uv_idx ------------------------------
__global__ void tet_emit(const int* __restrict__ tet, int F, int n,
                         const unsigned char* __restrict__ code,
                         const unsigned* __restrict__ off1, const unsigned* __restrict__ off2,
                         const unsigned* __restrict__ totals, const int* __restrict__ edgeMap,
                         float* __restrict__ out, long long nuvFloats) {
  int t = blockIdx.x * TPB + threadIdx.x;
  int c = (t < F) ? (int)code[t] : 0;
  int nt = c_NT[c];
  int m1 = (t < F) && (nt == 1);
  int m2 = (t < F) && (nt == 2);
  unsigned lane = threadIdx.x & 31u, wid = threadIdx.x >> 5;
  unsigned long long b1 = __ballot(m1), b2 = __ballot(m2);
  unsigned lm = (unsigned)((1ull << lane) - 1ull);
  unsigned p1 = (unsigned)__popcll(b1 & lm), p2 = (unsigned)__popcll(b2 & lm);
  __shared__ unsigned w1[TPB / 32], w2[TPB / 32];
  if (lane == 0) { w1[wid] = (unsigned)__popcll(b1); w2[wid] = (unsigned)__popcll(b2); }
  __syncthreads();
  unsigned wo1 = 0, wo2 = 0;
  for (unsigned i = 0; i < wid; ++i) { wo1 += w1[i]; wo2 += w2[i]; }
  unsigned r1 = off1[blockIdx.x] + wo1 + p1;
  unsigned r2 = off2[blockIdx.x] + wo2 + p2;

  unsigned Nv = totals[0], Nf1 = totals[1], Nm2 = totals[2];
  size_t facesBase = (size_t)Nv * 6;
  size_t nFaces = (size_t)Nf1 + 2ull * (size_t)Nm2;
  int* faces = (int*)(out + facesBase);
  int* uvidx = (int*)(out + facesBase + nFaces * 3 + (size_t)nuvFloats);

  if (m1 | m2) {
    const int4 vt = ((const int4*)tet)[t];
    int vv[4] = {vt.x, vt.y, vt.z, vt.w};
    int n2 = n * n;
    auto edgeVert = [&](int le) -> int {
      int a = vv[c_EC[le][0]], bb = vv[c_EC[le][1]];
      int u = a < bb ? a : bb, w = a < bb ? bb : a;
      int dlt = w - u;
      int dx = dlt >= n2; dlt -= dx * n2;
      int dy = dlt >= n;  dlt -= dy * n;
      int k = (dx * 4 + dy * 2 + dlt) - 1;
      return edgeMap[u * 7 + k];
    };
    if (m1) {
      size_t fr = (size_t)r1;
      #pragma unroll
      for (int j = 0; j < 3; ++j) faces[fr * 3 + j] = edgeVert(c_TRI[c][j]);
      uvidx[fr * 3 + 0] = t * 4; uvidx[fr * 3 + 1] = t * 4 + 1; uvidx[fr * 3 + 2] = t * 4 + 2;
    } else {
      size_t fr = (size_t)Nf1 + 2ull * (size_t)r2;
      #pragma unroll
      for (int j = 0; j < 3; ++j) faces[fr * 3 + j] = edgeVert(c_TRI[c][j]);
      #pragma unroll
      for (int j = 0; j < 3; ++j) faces[(fr + 1) * 3 + j] = edgeVert(c_TRI[c][3 + j]);
      uvidx[fr * 3 + 0] = t * 4;       uvidx[fr * 3 + 1] = t * 4 + 1; uvidx[fr * 3 + 2] = t * 4 + 2;
      uvidx[(fr + 1) * 3 + 0] = t * 4; uvidx[(fr + 1) * 3 + 1] = t * 4 + 2; uvidx[(fr + 1) * 3 + 2] = t * 4 + 3;
    }
  }
}

// ---------------- Pass 3: dense uv table (dominant bandwidth, NT stores) ----
__global__ void uv_fill(const unsigned* __restrict__ totals, int nuv,
                        float inv, float pad, float* __restrict__ out) {
  int i = blockIdx.x * TPB + threadIdx.x;        // within row: cell*4 + c
  int rowLen = nuv * 4;
  if (i >= rowLen) return;
  int iy = blockIdx.y;
  unsigned Nv = totals[0], Nf1 = totals[1], Nm2 = totals[2];
  size_t base = (size_t)Nv * 6 + ((size_t)Nf1 + 2ull * (size_t)Nm2) * 3;
  int ix = i >> 2, c = i & 3;
  float tx = (float)ix * inv, ty = (float)iy * inv;
  float ux = tx + ((c == 1 || c == 2) ? pad : 0.f);
  float uy = ty + ((c >= 2) ? pad : 0.f);
  size_t o = base + ((size_t)iy * rowLen + i) * 2;
  __builtin_nontemporal_store(ux, &out[o]);
  __builtin_nontemporal_store(uy, &out[o + 1]);
}

extern "C" void kernel_launch(void* const* d_in, const int* in_sizes, int n_in,
                              void* d_out, int out_size, void* d_ws, size_t ws_size,
                              hipStream_t stream) {
  const float* pos = (const float*)d_in[0];
  const float* sdf = (const float*)d_in[1];
  const int*   tet = (const int*)d_in[2];
  int N = in_sizes[1];
  int F = in_sizes[2] / 4;
  int n = (int)llround(cbrt((double)N));          // lattice side (R+1); n <= ZB
  int M = N * 7;                                  // edge slots
  int EB = n * n;                                 // one block per (x,y) column
  int TB = (F + TPB - 1) / TPB;
  long long halfIdx = (2ll * F + 1) / 2;          // (max_idx+1)//2
  int nuv = (int)ceil(sqrt((double)halfIdx));

  // carve workspace (all consumed data is written before read; deterministic)
  uint8_t* w = (uint8_t*)d_ws;
  auto carve = [&](size_t bytes) -> void* {
    void* p = (void*)w;
    w += (bytes + 255) & ~(size_t)255;
    return p;
  };
  unsigned char* tetCode = (unsigned char*)carve((size_t)F);
  unsigned* eSums  = (unsigned*)carve((size_t)EB * 4);
  unsigned* eOff   = (unsigned*)carve((size_t)EB * 4);
  unsigned* s1     = (unsigned*)carve((size_t)TB * 4);
  unsigned* s2     = (unsigned*)carve((size_t)TB * 4);
  unsigned* o1     = (unsigned*)carve((size_t)TB * 4);
  unsigned* o2     = (unsigned*)carve((size_t)TB * 4);
  unsigned* totals = (unsigned*)carve(16);
  int* edgeMap     = (int*)carve((size_t)M * 4);
  (void)ws_size; (void)out_size; (void)n_in;

  dim3 egrid(n, n);                               // blockIdx.x->y, blockIdx.y->x
  edge_classify<<<egrid, ZB, 0, stream>>>(sdf, n, eSums);
  scan_exclusive<<<1, TPB, 0, stream>>>(eSums, eOff, totals + 0, EB);   // Nv
  tet_classify<<<TB, TPB, 0, stream>>>(tet, sdf, F, tetCode, s1, s2);
  scan_exclusive<<<1, TPB, 0, stream>>>(s1, o1, totals + 1, TB);        // Nf1
  scan_exclusive<<<1, TPB, 0, stream>>>(s2, o2, totals + 2, TB);        // Nm2
  edge_emit<<<egrid, ZB, 0, stream>>>(sdf, pos, n, eOff, edgeMap, (float*)d_out);
  long long uvRows = (long long)nuv * nuv * 4;
  tet_emit<<<TB, TPB, 0, stream>>>(tet, F, n, tetCode, o1, o2, totals, edgeMap,
                                   (float*)d_out, uvRows * 2);
  dim3 ugrid((unsigned)((4 * nuv + TPB - 1) / TPB), (unsigned)nuv);
  uv_fill<<<ugrid, TPB, 0, stream>>>(totals, nuv, 1.0f / (float)nuv,
                                     0.9f / (float)nuv, (float*)d_out);
}